// tstcc_soft_41918880809203
// MI455X (gfx1250) — compile-verified
//
#include <hip/hip_runtime.h>

typedef __attribute__((ext_vector_type(2))) float v2f;
typedef __attribute__((ext_vector_type(8))) float v8f;

#define BATCH 64
#define KW 8
#define PADW 4

__device__ __forceinline__ float ninf() { return -__builtin_inff(); }

// async memory -> LDS copy of one f32 element per active lane (GVS mode):
// copies from global address saddr + gByteOff to LDS byte offset ldsByteOff; tracked by ASYNCcnt.
__device__ __forceinline__ void async_ld_f32(uint32_t ldsByteOff, int32_t gByteOff,
                                             const float* saddr) {
  asm volatile("global_load_async_to_lds_b32 %0, %1, %2"
               :: "v"(ldsByteOff), "v"(gByteOff), "s"(saddr) : "memory");
}
__device__ __forceinline__ void async_wait0() {
  asm volatile("s_wait_asynccnt 0" ::: "memory");
}

// ---------------- conv1d (pad=4) + deterministic per-(b,co) BN partial sums --
__global__ void conv_stats_kernel(const float* __restrict__ x,
                                  const float* __restrict__ w,
                                  float* __restrict__ y,
                                  float* __restrict__ partial,
                                  int Cin, int Cout, int Lin, int Lc) {
  int b  = blockIdx.x / Cout;
  int co = blockIdx.x % Cout;
  __shared__ float wsh[512];     // Cin*KW <= 64*8
  __shared__ float red[512];
  int nw = Cin * KW;
  for (int i = threadIdx.x; i < nw; i += blockDim.x) wsh[i] = w[co * nw + i];
  __syncthreads();
  const float* xb = x + (size_t)b * Cin * Lin;
  float* yb = y + ((size_t)b * Cout + co) * Lc;
  float s = 0.f, s2 = 0.f;
  for (int p = threadIdx.x; p < Lc; p += blockDim.x) {
    float acc = 0.f;
    for (int ci = 0; ci < Cin; ++ci) {
      const float* xr = xb + (size_t)ci * Lin;
      const float* wr = wsh + ci * KW;
#pragma unroll
      for (int k = 0; k < KW; ++k) {
        int xi = p + k - PADW;
        if (xi >= 0 && xi < Lin) acc += xr[xi] * wr[k];
      }
    }
    yb[p] = acc;
    s += acc; s2 += acc * acc;
  }
  red[threadIdx.x] = s;
  red[256 + threadIdx.x] = s2;
  __syncthreads();
  for (int st = 128; st > 0; st >>= 1) {
    if ((int)threadIdx.x < st) {
      red[threadIdx.x]       += red[threadIdx.x + st];
      red[256 + threadIdx.x] += red[256 + threadIdx.x + st];
    }
    __syncthreads();
  }
  if (threadIdx.x == 0) {
    partial[(b * Cout + co) * 2 + 0] = red[0];
    partial[(b * Cout + co) * 2 + 1] = red[256];
  }
}

// ---------------- finalize batch stats (fixed-order, deterministic) ---------
__global__ void bn_finalize_kernel(const float* __restrict__ partial,
                                   float* __restrict__ stats,
                                   int Cout, float invCount) {
  int co = threadIdx.x;
  if (co < Cout) {
    float s = 0.f, s2 = 0.f;
    for (int b = 0; b < BATCH; ++b) {
      s  += partial[(b * Cout + co) * 2 + 0];
      s2 += partial[(b * Cout + co) * 2 + 1];
    }
    float mean = s * invCount;
    float var = fmaxf(s2 * invCount - mean * mean, 0.f);
    stats[co] = mean;
    stats[Cout + co] = rsqrtf(var + 1e-5f);
  }
}

// ---------------- BN apply + ReLU + MaxPool(k=2,s=2,pad=1) -----------------
__global__ void bn_relu_pool_kernel(const float* __restrict__ y,
                                    const float* __restrict__ stats,
                                    const float* __restrict__ gamma,
                                    const float* __restrict__ beta,
                                    float* __restrict__ outp,
                                    int Cout, int Lc, int Lp) {
  int b  = blockIdx.x / Cout;
  int co = blockIdx.x % Cout;
  float mean = stats[co], rstd = stats[Cout + co];
  float gg = gamma[co], bb = beta[co];
  const float* yr = y + ((size_t)b * Cout + co) * Lc;
  float* orow = outp + ((size_t)b * Cout + co) * Lp;
  for (int p = threadIdx.x; p < Lp; p += blockDim.x) {
    float m = ninf();
#pragma unroll
    for (int q = 0; q < 2; ++q) {
      int pos = 2 * p - 1 + q;
      if (pos >= 0 && pos < Lc) {
        float v = (yr[pos] - mean) * rstd * gg + bb;
        m = fmaxf(m, fmaxf(v, 0.f));
      }
    }
    orow[p] = m;
  }
}

// ---------------- temporal contrastive loss (flash-style WMMA Gram) --------
// One block per batch b. z = concat(z1_b, z2_b) [2T x C] staged into LDS via
// CDNA5 async global->LDS copies. Gram tiles via v_wmma_f32_16x16x4_f32 with
// two independent accumulator chains; online log-softmax per row.
__global__ void temp_cl_kernel(const float* __restrict__ p1,
                               const float* __restrict__ p2,
                               float* __restrict__ lossOut,
                               int T, int C, int Tpad2, float sigma, float scale) {
  extern __shared__ float smem[];
  const int stride = C + 2;                  // bank-conflict padding, keeps 8B align
  float* zL  = smem;
  float* red = smem + (size_t)Tpad2 * stride;
  int b = blockIdx.x;
  int T2 = 2 * T;

  // ---- async stage: rows [0,T) from p1, [T,2T) from p2 (both [B][C][T]) ----
  uint32_t zbase = (uint32_t)(uintptr_t)(void*)zL;   // LDS byte offset of zL
  {
    const float* src = p1 + (size_t)b * C * T;
    for (int idx = threadIdx.x; idx < T * C; idx += blockDim.x) {
      int r = idx / C, c = idx - r * C;
      async_ld_f32(zbase + (uint32_t)(r * stride + c) * 4u, (c * T + r) * 4, src);
    }
  }
  {
    const float* src = p2 + (size_t)b * C * T;
    for (int idx = threadIdx.x; idx < T * C; idx += blockDim.x) {
      int r = idx / C, c = idx - r * C;
      async_ld_f32(zbase + (uint32_t)((T + r) * stride + c) * 4u, (c * T + r) * 4, src);
    }
  }
  for (int idx = threadIdx.x; idx < (Tpad2 - T2) * C; idx += blockDim.x) {
    int r = T2 + idx / C, c = idx % C;
    zL[r * stride + c] = 0.f;
  }
  async_wait0();
  __syncthreads();

  int lane = threadIdx.x & 31;
  int wave = threadIdx.x >> 5;
  int nWaves = blockDim.x >> 5;
  int ln = lane & 15, hi = lane >> 4;
  int nStrips = Tpad2 >> 4;
  float lacc = 0.f;

  for (int sIdx = wave; sIdx < nStrips; sIdx += nWaves) {
    int r0 = sIdx << 4;
    float mx[8], se[8], wsim[8], wsum[8];
#pragma unroll
    for (int v = 0; v < 8; ++v) { mx[v] = ninf(); se[v] = 0.f; wsim[v] = 0.f; wsum[v] = 0.f; }
    const float* arow = zL + (size_t)(r0 + ln) * stride + 2 * hi;
    for (int ct = 0; ct < nStrips; ++ct) {
      int c0 = ct << 4;
      const float* brow = zL + (size_t)(c0 + ln) * stride + 2 * hi;
      v8f acc0 = {}, acc1 = {};
      for (int k0 = 0; k0 < C; k0 += 8) {     // two independent XDL chains
        v2f a0 = *(const v2f*)(arow + k0);
        v2f b0 = *(const v2f*)(brow + k0);
        acc0 = __builtin_amdgcn_wmma_f32_16x16x4_f32(false, a0, false, b0,
                                                     (short)0, acc0, false, false);
        v2f a1 = *(const v2f*)(arow + k0 + 4);
        v2f b1 = *(const v2f*)(brow + k0 + 4);
        acc1 = __builtin_amdgcn_wmma_f32_16x16x4_f32(false, a1, false, b1,
                                                     (short)0, acc1, false, false);
      }
      int j = c0 + ln;
      int j1 = (j < T) ? j : j - T;
      bool jlow = (j < T);
#pragma unroll
      for (int v = 0; v < 8; ++v) {
        int i = r0 + v + (hi << 3);
        float xv = acc0[v] + acc1[v];
        float wgt = 0.f;
        bool valid = (j < T2) && (i < T2);
        if (valid) {
          int i1 = (i < T) ? i : i - T;
          if (((i < T) == jlow) && (i1 == j1)) {
            valid = false;                       // dropped diagonal
          } else {
            int d = (i1 > j1) ? (i1 - j1) : (j1 - i1);
            float m = 2.f / (1.f + __expf((float)d * sigma));
            wgt = (m < 1e-6f) ? 0.f : m;
          }
        }
        if (valid) {
          if (xv > mx[v]) { se[v] = se[v] * __expf(mx[v] - xv) + 1.f; mx[v] = xv; }
          else            { se[v] += __expf(xv - mx[v]); }
          wsim[v] += wgt * xv;
          wsum[v] += wgt;
        }
      }
    }
    // merge the 16 lanes of each half-wave (row-wise LSE + plain sums)
#pragma unroll
    for (int off = 1; off < 16; off <<= 1) {
#pragma unroll
      for (int v = 0; v < 8; ++v) {
        float om = __shfl_xor(mx[v], off, 32);
        float os = __shfl_xor(se[v], off, 32);
        float nm = fmaxf(mx[v], om);
        float c0v = (se[v] == 0.f) ? 0.f : se[v] * __expf(mx[v] - nm);
        float c1v = (os   == 0.f) ? 0.f : os   * __expf(om   - nm);
        se[v] = c0v + c1v;
        mx[v] = nm;
        wsim[v] += __shfl_xor(wsim[v], off, 32);
        wsum[v] += __shfl_xor(wsum[v], off, 32);
      }
    }
    if (ln == 0) {
#pragma unroll
      for (int v = 0; v < 8; ++v) {
        int i = r0 + v + (hi << 3);
        if (i < T2 && se[v] > 0.f) {
          float lse = mx[v] + __logf(se[v]);
          lacc += wsum[v] * lse - wsim[v];
        }
      }
    }
  }
  red[threadIdx.x] = lacc;
  __syncthreads();
  for (int st = 128; st > 0; st >>= 1) {
    if ((int)threadIdx.x < st) red[threadIdx.x] += red[threadIdx.x + st];
    __syncthreads();
  }
  if (threadIdx.x == 0) lossOut[b] = red[0] * scale;
}

// ---------------- instance contrastive loss (one block per timestep) -------
__global__ void inst_cl_kernel(const float* __restrict__ p1,
                               const float* __restrict__ p2,
                               const float* __restrict__ sl,
                               float* __restrict__ lossOut,
                               int T, int C, float scale) {
  extern __shared__ float smem[];
  const int stride = C + 2;
  float* zL  = smem;                       // [128][stride]
  float* red = smem + 128 * stride;
  int t = blockIdx.x;

  uint32_t zbase = (uint32_t)(uintptr_t)(void*)zL;
  {
    for (int idx = threadIdx.x; idx < 64 * C; idx += blockDim.x) {
      int r = idx / C, c = idx - r * C;
      async_ld_f32(zbase + (uint32_t)(r * stride + c) * 4u,
                   ((r * C + c) * T + t) * 4, p1);
    }
    for (int idx = threadIdx.x; idx < 64 * C; idx += blockDim.x) {
      int r = idx / C, c = idx - r * C;
      async_ld_f32(zbase + (uint32_t)((64 + r) * stride + c) * 4u,
                   ((r * C + c) * T + t) * 4, p2);
    }
  }
  async_wait0();
  __syncthreads();

  int lane = threadIdx.x & 31;
  int wave = threadIdx.x >> 5;
  int nWaves = blockDim.x >> 5;
  int ln = lane & 15, hi = lane >> 4;
  float lacc = 0.f;

  for (int sIdx = wave; sIdx < 8; sIdx += nWaves) {   // 128 rows = 8 strips
    int r0 = sIdx << 4;
    float mx[8], se[8], wsim[8], wsum[8];
#pragma unroll
    for (int v = 0; v < 8; ++v) { mx[v] = ninf(); se[v] = 0.f; wsim[v] = 0.f; wsum[v] = 0.f; }
    const float* arow = zL + (size_t)(r0 + ln) * stride + 2 * hi;
    for (int ct = 0; ct < 8; ++ct) {
      int c0 = ct << 4;
      const float* brow = zL + (size_t)(c0 + ln) * stride + 2 * hi;
      v8f acc0 = {}, acc1 = {};
      for (int k0 = 0; k0 < C; k0 += 8) {
        v2f a0 = *(const v2f*)(arow + k0);
        v2f b0 = *(const v2f*)(brow + k0);
        acc0 = __builtin_amdgcn_wmma_f32_16x16x4_f32(false, a0, false, b0,
                                                     (short)0, acc0, false, false);
        v2f a1 = *(const v2f*)(arow + k0 + 4);
        v2f b1 = *(const v2f*)(brow + k0 + 4);
        acc1 = __builtin_amdgcn_wmma_f32_16x16x4_f32(false, a1, false, b1,
                                                     (short)0, acc1, false, false);
      }
      int j = c0 + ln;
      int j1 = j & 63;
      bool jlow = (j < 64);
#pragma unroll
      for (int v = 0; v < 8; ++v) {
        int i = r0 + v + (hi << 3);
        int i1 = i & 63;
        float xv = acc0[v] + acc1[v];
        bool excl = ((i < 64) == jlow) && (i1 == j1);
        if (!excl) {
          float wgt = sl[i1 * 64 + j1];
          if (xv > mx[v]) { se[v] = se[v] * __expf(mx[v] - xv) + 1.f; mx[v] = xv; }
          else            { se[v] += __expf(xv - mx[v]); }
          wsim[v] += wgt * xv;
          wsum[v] += wgt;
        }
      }
    }
#pragma unroll
    for (int off = 1; off < 16; off <<= 1) {
#pragma unroll
      for (int v = 0; v < 8; ++v) {
        float om = __shfl_xor(mx[v], off, 32);
        float os = __shfl_xor(se[v], off, 32);
        float nm = fmaxf(mx[v], om);
        float c0v = (se[v] == 0.f) ? 0.f : se[v] * __expf(mx[v] - nm);
        float c1v = (os   == 0.f) ? 0.f : os   * __expf(om   - nm);
        se[v] = c0v + c1v;
        mx[v] = nm;
        wsim[v] += __shfl_xor(wsim[v], off, 32);
        wsum[v] += __shfl_xor(wsum[v], off, 32);
      }
    }
    if (ln == 0) {
#pragma unroll
      for (int v = 0; v < 8; ++v) {
        float lse = mx[v] + __logf(se[v]);
        lacc += wsum[v] * lse - wsim[v];
      }
    }
  }
  red[threadIdx.x] = lacc;
  __syncthreads();
  for (int st = 128; st > 0; st >>= 1) {
    if ((int)threadIdx.x < st) red[threadIdx.x] += red[threadIdx.x + st];
    __syncthreads();
  }
  if (threadIdx.x == 0) lossOut[t] = red[0] * scale;
}

// ---------------- final linear head: [64,16640] @ [16640,5] + bias ---------
__global__ void classifier_kernel(const float* __restrict__ a,
                                  const float* __restrict__ wf,
                                  const float* __restrict__ bf,
                                  float* __restrict__ lg) {
  int b = blockIdx.x;
  __shared__ float red[5][256];
  float acc[5] = {0.f, 0.f, 0.f, 0.f, 0.f};
  const float* ar = a + (size_t)b * 16640;
  for (int f = threadIdx.x; f < 16640; f += blockDim.x) {
    float av = ar[f];
    const float* wr = wf + (size_t)f * 5;
#pragma unroll
    for (int c = 0; c < 5; ++c) acc[c] += av * wr[c];
  }
#pragma unroll
  for (int c = 0; c < 5; ++c) red[c][threadIdx.x] = acc[c];
  __syncthreads();
  for (int st = 128; st > 0; st >>= 1) {
    if ((int)threadIdx.x < st)
#pragma unroll
      for (int c = 0; c < 5; ++c) red[c][threadIdx.x] += red[c][threadIdx.x + st];
    __syncthreads();
  }
  if (threadIdx.x < 5) lg[b * 5 + threadIdx.x] = red[threadIdx.x][0] + bf[threadIdx.x];
}

// ---------------- deterministic total-loss reduction -----------------------
__global__ void loss_sum_kernel(const float* __restrict__ lp, int n,
                                float* __restrict__ outv) {
  __shared__ float red[256];
  float s = 0.f;
  for (int i = threadIdx.x; i < n; i += blockDim.x) s += lp[i];
  red[threadIdx.x] = s;
  __syncthreads();
  for (int st = 128; st > 0; st >>= 1) {
    if ((int)threadIdx.x < st) red[threadIdx.x] += red[threadIdx.x + st];
    __syncthreads();
  }
  if (threadIdx.x == 0) *outv = red[0];
}

extern "C" void kernel_launch(void* const* d_in, const int* in_sizes, int n_in,
                              void* d_out, int out_size, void* d_ws, size_t ws_size,
                              hipStream_t stream) {
  const float* aug1 = (const float*)d_in[0];
  const float* aug2 = (const float*)d_in[1];
  const float* sl   = (const float*)d_in[2];
  const float* w[3]  = {(const float*)d_in[3], (const float*)d_in[6], (const float*)d_in[9]};
  const float* gm[3] = {(const float*)d_in[4], (const float*)d_in[7], (const float*)d_in[10]};
  const float* bt[3] = {(const float*)d_in[5], (const float*)d_in[8], (const float*)d_in[11]};
  const float* wf = (const float*)d_in[12];
  const float* bf = (const float*)d_in[13];
  float* out = (float*)d_out;
  float* ws  = (float*)d_ws;

  const int Cin[3] = {1, 32, 64};
  const int Co[3]  = {32, 64, 128};
  const int Lin[3] = {1024, 513, 258};
  const int Lc[3]  = {1025, 514, 259};
  const int Lp[3]  = {513, 258, 130};
  const float sig[3] = {2.f, 4.f, 8.f};   // TAU * 2^d

  // workspace layout (floats); total ~25.5 MB
  const size_t CONVMAX = 2121728;   // 64*128*259
  const size_t POOLMAX = 1056768;   // 64*64*258
  float* convb = ws;
  float* part  = ws + CONVMAX;      // 16384
  float* stats = part + 16384;      // 256
  float* lossP = stats + 256;       // 1093 slots (rounded 1100)
  float* pbase = lossP + 1100;
  float* pA0 = pbase;
  float* pB0 = pA0 + POOLMAX;
  float* pA1 = pB0 + POOLMAX;
  float* pB1 = pA1 + POOLMAX;

  const int A1OFF = 640;
  const int A2OFF = 640 + 1064960;
  const int LOSSOFF = 640 + 2 * 1064960;

  float* so1[3] = {pA0, pA1, out + A1OFF};
  float* so2[3] = {pB0, pB1, out + A2OFF};
  const int tOff[3] = {0, 64, 128};
  const int iOff[3] = {192, 192 + 513, 192 + 513 + 258};

  const float* in1 = aug1;
  const float* in2 = aug2;
  for (int s = 0; s < 3; ++s) {
    dim3 gc(BATCH * Co[s]);
    float invCnt = 1.f / (64.f * (float)Lc[s]);

    conv_stats_kernel<<<gc, 256, 0, stream>>>(in1, w[s], convb, part, Cin[s], Co[s], Lin[s], Lc[s]);
    bn_finalize_kernel<<<1, 128, 0, stream>>>(part, stats, Co[s], invCnt);
    bn_relu_pool_kernel<<<gc, 256, 0, stream>>>(convb, stats, gm[s], bt[s], so1[s], Co[s], Lc[s], Lp[s]);

    conv_stats_kernel<<<gc, 256, 0, stream>>>(in2, w[s], convb, part, Cin[s], Co[s], Lin[s], Lc[s]);
    bn_finalize_kernel<<<1, 128, 0, stream>>>(part, stats, Co[s], invCnt);
    bn_relu_pool_kernel<<<gc, 256, 0, stream>>>(convb, stats, gm[s], bt[s], so2[s], Co[s], Lc[s], Lp[s]);

    int T = Lp[s], C = Co[s];
    int Tpad2 = ((2 * T + 15) / 16) * 16;
    float scale = 0.5f / (2.f * 64.f * (float)T);   // (1-LAM)=LAM=0.5 over 2*b*t
    size_t smemT = (size_t)(Tpad2 * (C + 2) + 256) * sizeof(float);
    temp_cl_kernel<<<64, 256, smemT, stream>>>(so1[s], so2[s], lossP + tOff[s],
                                               T, C, Tpad2, sig[s], scale);
    size_t smemI = (size_t)(128 * (C + 2) + 256) * sizeof(float);
    inst_cl_kernel<<<T, 256, smemI, stream>>>(so1[s], so2[s], sl, lossP + iOff[s],
                                              T, C, scale);
    in1 = so1[s];
    in2 = so2[s];
  }

  classifier_kernel<<<64, 256, 0, stream>>>(out + A1OFF, wf, bf, out);
  classifier_kernel<<<64, 256, 0, stream>>>(out + A2OFF, wf, bf, out + 320);
  loss_sum_kernel<<<1, 256, 0, stream>>>(lossP, 1093, out + LOSSOFF);

  (void)in_sizes; (void)n_in; (void)out_size; (void)ws_size;
}